// GAT_5523327942737
// MI455X (gfx1250) — compile-verified
//
#include <hip/hip_runtime.h>
#include <math.h>

// ---------------------------------------------------------------------------
// Types for CDNA5 WMMA (wave32). f32 16x16x4: A/B = 2 VGPRs/lane, C/D = 8.
// ---------------------------------------------------------------------------
typedef __attribute__((ext_vector_type(2))) float v2f;
typedef __attribute__((ext_vector_type(8))) float v8f;

#define NEG_SLOPE 0.2f

// ---------------------------------------------------------------------------
// Dense GEMM: Hout[N,F] = X[N,K] @ W[K,F], F = TILES*16, K % 4 == 0, N % 16 == 0.
// One wave (blockDim=32) computes a 16-row x F-column stripe.
// A-fragment (16x4 f32) layout per ISA 7.12.2:
//   lane 0..15  : M = lane,    VGPR0 = K0,   VGPR1 = K0+1
//   lane 16..31 : M = lane-16, VGPR0 = K0+2, VGPR1 = K0+3
// B-fragment (4x16) mirrors with N striped across lanes.
// C/D (16x16 f32): VGPR r -> M = r + 8*(lane>=16), N = lane&15.
// ---------------------------------------------------------------------------
template <int TILES>
__global__ __launch_bounds__(32)
void gat_gemm_wmma(const float* __restrict__ X, const float* __restrict__ W,
                   float* __restrict__ Hout, int N, int K) {
  const int F    = TILES * 16;
  const int row0 = blockIdx.x * 16;
  const int lane = threadIdx.x;          // 0..31 (wave32)
  const int l15  = lane & 15;
  const int kh   = (lane >> 4) * 2;      // 0 or 2: which K-pair this half-wave holds
  const int m    = row0 + l15;

  v8f acc[TILES];
#pragma unroll
  for (int t = 0; t < TILES; ++t) acc[t] = (v8f){0.f,0.f,0.f,0.f,0.f,0.f,0.f,0.f};

  for (int k0 = 0; k0 < K; k0 += 4) {
    v2f a;
    a.x = X[(long)m * K + k0 + kh];
    a.y = X[(long)m * K + k0 + kh + 1];
#pragma unroll
    for (int t = 0; t < TILES; ++t) {
      v2f b;
      b.x = W[(long)(k0 + kh)     * F + t * 16 + l15];
      b.y = W[(long)(k0 + kh + 1) * F + t * 16 + l15];
      acc[t] = __builtin_amdgcn_wmma_f32_16x16x4_f32(
          /*neg_a=*/false, a, /*neg_b=*/false, b,
          /*c_mod=*/(short)0, acc[t], /*reuse_a=*/false, /*reuse_b=*/false);
    }
  }

  const int hi = (lane >> 4) * 8;
#pragma unroll
  for (int t = 0; t < TILES; ++t)
#pragma unroll
    for (int r = 0; r < 8; ++r)
      Hout[(long)(row0 + hi + r) * F + t * 16 + l15] = acc[t][r];
}

// ---------------------------------------------------------------------------
// Per-node attention logits: as[n] = h[n,:].a_src, ad[n] = h[n,:].a_dst
// One wave per node, shfl tree reduction (wave32).
// ---------------------------------------------------------------------------
__global__ void gat_alpha(const float* __restrict__ h,
                          const float* __restrict__ a_src,
                          const float* __restrict__ a_dst,
                          float* __restrict__ as, float* __restrict__ ad,
                          int N, int C) {
  int wid  = (int)((blockIdx.x * (long)blockDim.x + threadIdx.x) >> 5);
  int lane = threadIdx.x & 31;
  if (wid >= N) return;
  float ss = 0.f, sd = 0.f;
  for (int c = lane; c < C; c += 32) {
    float hv = h[(long)wid * C + c];
    ss += hv * a_src[c];
    sd += hv * a_dst[c];
  }
#pragma unroll
  for (int off = 16; off > 0; off >>= 1) {
    ss += __shfl_down(ss, off, 32);
    sd += __shfl_down(sd, off, 32);
  }
  if (lane == 0) { as[wid] = ss; ad[wid] = sd; }
}

// ---------------------------------------------------------------------------
// Edge pass 1: e = leaky_relu(as[src]+ad[dst]); segment max via float atomicMax.
// Edge ids >= E are self-loops (src=dst=id-E), matching the reference concat.
// ---------------------------------------------------------------------------
__global__ void gat_edge_logits(const int* __restrict__ src,
                                const int* __restrict__ dst,
                                const float* __restrict__ as,
                                const float* __restrict__ ad,
                                float* __restrict__ ebuf,
                                float* __restrict__ mbuf,
                                int E, int N) {
  int e = blockIdx.x * blockDim.x + threadIdx.x;
  int ET = E + N;
  if (e >= ET) return;
  int s = (e < E) ? src[e] : (e - E);
  int d = (e < E) ? dst[e] : (e - E);
  float v = as[s] + ad[d];
  v = (v > 0.f) ? v : NEG_SLOPE * v;
  ebuf[e] = v;
  atomicMax(&mbuf[d], v);   // -> global_atomic_max_num_f32
}

// Edge pass 2: ex = exp(e - m[dst]); den[dst] += ex; ebuf overwritten with ex.
__global__ void gat_edge_exp(const int* __restrict__ dst,
                             float* __restrict__ ebuf,
                             const float* __restrict__ mbuf,
                             float* __restrict__ den,
                             int E, int N) {
  int e = blockIdx.x * blockDim.x + threadIdx.x;
  int ET = E + N;
  if (e >= ET) return;
  int d = (e < E) ? dst[e] : (e - E);
  float ex = __expf(ebuf[e] - mbuf[d]);
  ebuf[e] = ex;
  atomicAdd(&den[d], ex);
}

// Edge pass 3: agg[dst,:] += h[src,:] * (ex / (den[dst]+1e-16)).
// One wave per edge; lanes stride channels -> coalesced gather + atomic scatter.
__global__ void gat_edge_aggregate(const int* __restrict__ src,
                                   const int* __restrict__ dst,
                                   const float* __restrict__ ebuf,
                                   const float* __restrict__ den,
                                   const float* __restrict__ h,
                                   float* __restrict__ agg,
                                   int E, int N, int C) {
  int wid  = (int)((blockIdx.x * (long)blockDim.x + threadIdx.x) >> 5);
  int lane = threadIdx.x & 31;
  int ET = E + N;
  if (wid >= ET) return;
  int s = (wid < E) ? src[wid] : (wid - E);
  int d = (wid < E) ? dst[wid] : (wid - E);
  float alpha = ebuf[wid] / (den[d] + 1e-16f);
  for (int c = lane; c < C; c += 32)
    atomicAdd(&agg[(long)d * C + c], h[(long)s * C + c] * alpha);
}

// Epilogue: out = (agg + bias) [+ ReLU]
__global__ void gat_bias_act(const float* __restrict__ agg,
                             const float* __restrict__ bias,
                             float* __restrict__ out,
                             int N, int C, int do_relu) {
  int i = blockIdx.x * blockDim.x + threadIdx.x;
  if (i >= N * C) return;
  float v = agg[i] + bias[i % C];
  if (do_relu) v = fmaxf(v, 0.f);
  out[i] = v;
}

__global__ void gat_fill(float* __restrict__ p, float v, int n) {
  int i = blockIdx.x * blockDim.x + threadIdx.x;
  if (i < n) p[i] = v;
}

// ---------------------------------------------------------------------------
// Host launcher
// ---------------------------------------------------------------------------
extern "C" void kernel_launch(void* const* d_in, const int* in_sizes, int n_in,
                              void* d_out, int out_size, void* d_ws, size_t ws_size,
                              hipStream_t stream) {
  const float* fea = (const float*)d_in[0];   // [B, N, 128]
  const int*   ei  = (const int*)  d_in[1];   // [B, 2, E]
  const float* W1  = (const float*)d_in[2];   // [128, 128]
  const float* av1s= (const float*)d_in[3];   // [1, 128]
  const float* av1d= (const float*)d_in[4];   // [1, 128]
  const float* b1  = (const float*)d_in[5];   // [128]
  const float* W2  = (const float*)d_in[6];   // [128, 64]
  const float* av2s= (const float*)d_in[7];   // [1, 64]
  const float* av2d= (const float*)d_in[8];   // [1, 64]
  const float* b2  = (const float*)d_in[9];   // [64]
  float* out = (float*)d_out;

  const int B  = 2;
  const int C1 = 128, C2 = 64;
  const int N  = in_sizes[0] / (B * C1);      // 50000
  const int E  = in_sizes[1] / (B * 2);       // 800000
  const int ET = E + N;                       // self-loops appended

  // Workspace carve-out (floats)
  float* w   = (float*)d_ws;
  float* h1   = w; w += (long)N * C1;   // layer-1 features / activated output
  float* agg1 = w; w += (long)N * C1;
  float* h2   = w; w += (long)N * C2;
  float* agg2 = w; w += (long)N * C2;
  float* asb  = w; w += N;
  float* adb  = w; w += N;
  float* mb   = w; w += N;
  float* denb = w; w += N;
  float* ebuf = w; w += ET;

  const int TB = 256;
  dim3 blk(TB);
  int gN       = (N + TB - 1) / TB;
  int gNC1     = (N * C1 + TB - 1) / TB;
  int gNC2     = (N * C2 + TB - 1) / TB;
  int gEdges   = (ET + TB - 1) / TB;
  int gEdgeWav = (ET * 32 + TB - 1) / TB;     // wave per edge
  int gNodeWav = (N * 32 + TB - 1) / TB;      // wave per node
  int gGemm    = N / 16;                      // N=50000 is 16-aligned

  for (int b = 0; b < B; ++b) {
    const float* xb   = fea + (long)b * N * C1;
    const int*   srcb = ei + (long)b * 2 * E;
    const int*   dstb = srcb + E;
    float*       outb = out + (long)b * N * C2;

    // ---------------- Layer 1 (d=128 -> 128, ReLU) ----------------
    gat_fill<<<gN,   blk, 0, stream>>>(mb,  -INFINITY, N);
    gat_fill<<<gN,   blk, 0, stream>>>(denb, 0.f, N);
    gat_fill<<<gNC1, blk, 0, stream>>>(agg1, 0.f, N * C1);

    gat_gemm_wmma<8><<<gGemm, dim3(32), 0, stream>>>(xb, W1, h1, N, C1);
    gat_alpha<<<gNodeWav, blk, 0, stream>>>(h1, av1s, av1d, asb, adb, N, C1);
    gat_edge_logits<<<gEdges, blk, 0, stream>>>(srcb, dstb, asb, adb, ebuf, mb, E, N);
    gat_edge_exp<<<gEdges, blk, 0, stream>>>(dstb, ebuf, mb, denb, E, N);
    gat_edge_aggregate<<<gEdgeWav, blk, 0, stream>>>(srcb, dstb, ebuf, denb, h1, agg1, E, N, C1);
    gat_bias_act<<<gNC1, blk, 0, stream>>>(agg1, b1, h1, N, C1, /*relu=*/1);

    // ---------------- Layer 2 (128 -> 64) ----------------
    gat_fill<<<gN,   blk, 0, stream>>>(mb,  -INFINITY, N);
    gat_fill<<<gN,   blk, 0, stream>>>(denb, 0.f, N);
    gat_fill<<<gNC2, blk, 0, stream>>>(agg2, 0.f, N * C2);

    gat_gemm_wmma<4><<<gGemm, dim3(32), 0, stream>>>(h1, W2, h2, N, C1);
    gat_alpha<<<gNodeWav, blk, 0, stream>>>(h2, av2s, av2d, asb, adb, N, C2);
    gat_edge_logits<<<gEdges, blk, 0, stream>>>(srcb, dstb, asb, adb, ebuf, mb, E, N);
    gat_edge_exp<<<gEdges, blk, 0, stream>>>(dstb, ebuf, mb, denb, E, N);
    gat_edge_aggregate<<<gEdgeWav, blk, 0, stream>>>(srcb, dstb, ebuf, denb, h2, agg2, E, N, C2);
    gat_bias_act<<<gNC2, blk, 0, stream>>>(agg2, b2, outb, N, C2, /*relu=*/0);
  }
}